// ConnectedComponentCriterion_47459388621514
// MI455X (gfx1250) — compile-verified
//
#include <hip/hip_runtime.h>
#include <stdint.h>
#include <limits.h>

#define H_DIM 128
#define W_DIM 128
#define HW    (H_DIM * W_DIM)          // 16384
#define BIG   HW                        // background label / sentinel
#define NTHREADS 256

typedef __attribute__((ext_vector_type(4))) unsigned int v4u;
typedef __attribute__((ext_vector_type(8))) int          v8i;
typedef __attribute__((ext_vector_type(4))) int          v4i;

// -------- dynamic LDS layout (bytes) --------
// [0,      65536)  float Mt[16384]        mask tile (filled by TDM)
// [65536,  131076) int   L[16385]         labels (+1 entry for BIG pointer-jump)
// [131076, 196616) int   A[16385]         per-segment area histogram
// [196616, 200712) int   RED[256*4]       per-thread top-2 candidates
// [200712, 201736) float FR[256]          float reduction scratch
// [201736, 201768) int   MISC[8]          {changed, j, have2, rmin, rmax, cmin, cmax}
#define OFF_MT   0
#define OFF_L    65536
#define OFF_A    131076
#define OFF_RED  196616
#define OFF_FR   200712
#define OFF_MISC 201736
#define SHMEM_BYTES 201792

__device__ __forceinline__ bool better(int a, int i, int b, int j) {
    // jax top_k order: larger value first; ties -> smaller index first
    return (a > b) || (a == b && i < j);
}

__global__ __launch_bounds__(NTHREADS)
void cc_mask_kernel(const float* __restrict__ masks, float* __restrict__ losses)
{
    extern __shared__ char smem[];
    float* Mt   = (float*)(smem + OFF_MT);
    int*   L    = (int*)  (smem + OFF_L);
    int*   A    = (int*)  (smem + OFF_A);
    int*   RED  = (int*)  (smem + OFF_RED);
    float* FR   = (float*)(smem + OFF_FR);
    int*   MISC = (int*)  (smem + OFF_MISC);

    const int tid = threadIdx.x;
    const int b   = blockIdx.x;
    const float* __restrict__ src = masks + (size_t)b * HW;

    // ---------- Phase 0: bulk tile load global -> LDS via Tensor Data Mover ----------
#if defined(__gfx1250__) && __has_builtin(__builtin_amdgcn_tensor_load_to_lds)
    if (tid < 32) {  // wave-uniform: only wave 0 issues the DMA (EXEC ignored by TDM)
        // Generic LDS pointer: low 32 bits == LDS byte offset (aperture in high bits).
        uint32_t lds_addr = (uint32_t)(uintptr_t)(void*)Mt;
        uint64_t ga       = (uint64_t)(uintptr_t)(const void*)src;
        // D# group 0: count=1 | lds_addr | global_addr[56:0] | type=2 ("image")
        v4u g0 = { 1u,
                   lds_addr,
                   (uint32_t)ga,
                   (uint32_t)((ga >> 32) & 0x01FFFFFFu) | (2u << 30) };
        // D# group 1: data_size=2 (4B) ; tensor 128x128 f32 ; tile 128x128 ; stride0=128
        v8i g1 = { (int)(2u << 16),                 // wg_mask=0, data_size=4B
                   (int)(128u << 16),               // tensor_dim0[15:0] in hi half
                   (int)((0u) | (128u << 16)),      // tensor_dim0[31:16]=0 | tensor_dim1[15:0]
                   (int)((0u) | (128u << 16)),      // tensor_dim1[31:16]=0 | tile_dim0=128
                   (int)(128u),                     // tile_dim1=128 | tile_dim2=0
                   (int)(128u),                     // tensor_dim0_stride[31:0]=128
                   0,                               // stride hi | dim1_stride lo (unused, 2D)
                   0 };
        v4i gz4 = { 0, 0, 0, 0 };                   // groups 2/3 unused for 2D tile
        v8i gz8 = { 0, 0, 0, 0, 0, 0, 0, 0 };
        __builtin_amdgcn_tensor_load_to_lds(g0, g1, gz4, gz4, gz8, 0);
        __builtin_amdgcn_s_wait_tensorcnt(0);       // s_wait_tensorcnt 0
    }
#else
    for (int i = tid; i < HW; i += NTHREADS) Mt[i] = src[i];
#endif
    __syncthreads();

    // ---------- Phase 1: init labels ----------
    for (int i = tid; i < HW; i += NTHREADS)
        L[i] = (Mt[i] > 0.0f) ? i : BIG;
    if (tid == 0) L[HW] = BIG;   // pointer-jump target for BIG
    __syncthreads();

    // ---------- Phase 2: min-label propagation + pointer jumping ----------
    // Labels are monotone non-increasing; any update order converges to the
    // unique fixpoint (min flat index per 8-connected component), which is
    // exactly what the reference's synchronous loop converges to.
    for (;;) {
        if (tid == 0) MISC[0] = 0;
        __syncthreads();                       // (a)
        int ch = 0;
        for (int i = tid; i < HW; i += NTHREADS) {
            int li = L[i];
            if (li == BIG) continue;           // background never updates
            int r = i >> 7, c = i & (W_DIM - 1);
            int m = li;
            if (r > 0) {
                int base = i - W_DIM;
                if (c > 0)         m = min(m, L[base - 1]);
                                   m = min(m, L[base]);
                if (c < W_DIM - 1) m = min(m, L[base + 1]);
            }
            if (c > 0)             m = min(m, L[i - 1]);
            if (c < W_DIM - 1)     m = min(m, L[i + 1]);
            if (r < H_DIM - 1) {
                int base = i + W_DIM;
                if (c > 0)         m = min(m, L[base - 1]);
                                   m = min(m, L[base]);
                if (c < W_DIM - 1) m = min(m, L[base + 1]);
            }
            m = L[m];                          // pointer jump 1
            m = L[m];                          // pointer jump 2
            if (m < li) { L[i] = m; ch = 1; }
        }
        if (ch) MISC[0] = 1;
        __syncthreads();                       // (b)
        if (MISC[0] == 0) break;               // uniform across block
        __syncthreads();                       // (c) protect reset vs. reads
    }

    // ---------- Phase 3: per-segment areas (LDS ds_add_u32 atomics) ----------
    for (int i = tid; i <= HW; i += NTHREADS) A[i] = 0;
    __syncthreads();
    for (int i = tid; i < HW; i += NTHREADS)
        atomicAdd(&A[L[i]], 1);
    __syncthreads();

    // ---------- top-2 areas, jax top_k semantics (a = area>0 ? area : -1) ----------
    int a1 = -1, i1 = HW + 1, a2 = -1, i2 = HW + 1;
    for (int i = tid; i <= HW; i += NTHREADS) {
        int av = A[i];
        int a  = (av > 0) ? av : -1;
        if (better(a, i, a1, i1))      { a2 = a1; i2 = i1; a1 = a; i1 = i; }
        else if (better(a, i, a2, i2)) { a2 = a;  i2 = i; }
    }
    RED[tid * 4 + 0] = a1; RED[tid * 4 + 1] = i1;
    RED[tid * 4 + 2] = a2; RED[tid * 4 + 3] = i2;
    __syncthreads();
    if (tid == 0) {
        int A1 = RED[0], I1 = RED[1], A2 = RED[2], I2 = RED[3];
        for (int t = 1; t < NTHREADS; ++t) {
            int qa1 = RED[t * 4], qi1 = RED[t * 4 + 1];
            int qa2 = RED[t * 4 + 2], qi2 = RED[t * 4 + 3];
            if (better(A1, I1, qa1, qi1)) {
                if (better(qa1, qi1, A2, I2)) { A2 = qa1; I2 = qi1; }
            } else {
                int na2, ni2;
                if (better(A1, I1, qa2, qi2)) { na2 = A1; ni2 = I1; }
                else                          { na2 = qa2; ni2 = qi2; }
                A1 = qa1; I1 = qi1; A2 = na2; I2 = ni2;
            }
        }
        MISC[1] = I2;                 // j  (second-largest component; may be BIG = background)
        MISC[2] = (A2 > 0) ? 1 : 0;   // have2
        MISC[3] = INT_MAX; MISC[4] = -1; MISC[5] = INT_MAX; MISC[6] = -1;
    }
    __syncthreads();

    // ---------- Phase 4: bounding box of component j (ds_min/max_i32) ----------
    const int j     = MISC[1];
    const int have2 = MISC[2];
    if (have2) {
        for (int i = tid; i < HW; i += NTHREADS) {
            if (L[i] == j) {
                int r = i >> 7, c = i & (W_DIM - 1);
                atomicMin(&MISC[3], r);
                atomicMax(&MISC[4], r);
                atomicMin(&MISC[5], c);
                atomicMax(&MISC[6], c);
            }
        }
    }
    __syncthreads();

    // ---------- Phase 5: masked mean (deterministic tree reduction) ----------
    const int r0 = MISC[3], r1 = MISC[4], c0 = MISC[5], c1 = MISC[6];
    float s = 0.0f;
    for (int i = tid; i < HW; i += NTHREADS) {
        int r = i >> 7, c = i & (W_DIM - 1);
        bool inbox = have2 && (r >= r0) && (r <= r1) && (c >= c0) && (c <= c1);
        if (!inbox) s += Mt[i];
    }
    FR[tid] = s;
    __syncthreads();
    for (int off = NTHREADS / 2; off > 0; off >>= 1) {
        if (tid < off) FR[tid] += FR[tid + off];
        __syncthreads();
    }
    if (tid == 0) losses[b] = FR[0] * (1.0f / (float)HW);
}

// Fixed-order final reduction: bitwise deterministic across graph replays.
__global__ void cc_reduce_kernel(const float* __restrict__ losses,
                                 float* __restrict__ out, int n)
{
    if (threadIdx.x == 0 && blockIdx.x == 0) {
        float s = 0.0f;
        for (int i = 0; i < n; ++i) s += losses[i];
        out[0] = (n > 0) ? (s / (float)n) : 0.0f;
    }
}

extern "C" void kernel_launch(void* const* d_in, const int* in_sizes, int n_in,
                              void* d_out, int out_size, void* d_ws, size_t ws_size,
                              hipStream_t stream)
{
    const float* masks  = (const float*)d_in[0];     // (8,16,128,128) f32
    float*       losses = (float*)d_ws;              // one loss per mask
    float*       outp   = (float*)d_out;             // scalar f32
    const int nMasks = in_sizes[0] / HW;             // 128

    cc_mask_kernel<<<nMasks, NTHREADS, SHMEM_BYTES, stream>>>(masks, losses);
    cc_reduce_kernel<<<1, 32, 0, stream>>>(losses, outp, nMasks);
}